// FlexibleSHREDDecoder_32195074851272
// MI455X (gfx1250) — compile-verified
//
#include <hip/hip_runtime.h>
#include <hip/hip_bf16.h>
#include <math.h>

typedef __attribute__((ext_vector_type(2))) float v2f;
typedef __attribute__((ext_vector_type(8))) float v8f;

#define B_  32
#define T_  512
#define D_  64
#define H_  64
#define G4H 256   // 4*H
#define S_  4096
#define BT  16384 // B*T

__device__ __forceinline__ v8f wmma4(v2f a, v2f b, v8f c) {
  // V_WMMA_F32_16X16X4_F32 : full-precision f32 matrix core op (wave32)
  return __builtin_amdgcn_wmma_f32_16x16x4_f32(false, a, false, b, (short)0, c,
                                               false, false);
}

__device__ __forceinline__ float sigf(float x) { return 1.0f / (1.0f + __expf(-x)); }
__device__ __forceinline__ float gelu_tanh(float x) {
  float x3 = x * x * x;
  return 0.5f * x * (1.0f + tanhf(0.7978845608028654f * (x + 0.044715f * x3)));
}

// ---------------------------------------------------------------------------
// Kernel 1: xg = enc @ Wx   [16384,64] x [64,256] -> [16384,256]
// One workgroup (8 waves) per 16-row M tile; each wave does two 16x16 N tiles.
// ---------------------------------------------------------------------------
__global__ __launch_bounds__(256) void k_xgemm(const float* __restrict__ enc,
                                               const float* __restrict__ Wx,
                                               float* __restrict__ xg) {
  __shared__ float sA[16 * 64];
  const int m0 = blockIdx.x * 16;
  for (int i = threadIdx.x; i < 16 * 64; i += 256)
    sA[i] = enc[(m0 + (i >> 6)) * 64 + (i & 63)];
  __syncthreads();

  const int wave = threadIdx.x >> 5;
  const int lane = threadIdx.x & 31;
  const int lo = lane & 15;
  const int hi = lane >> 4;
  const int koff = hi * 2;

  for (int ti = 0; ti < 2; ++ti) {
    const int n0 = (wave * 2 + ti) * 16;
    v8f acc = {};
    for (int kb = 0; kb < 64; kb += 4) {
      v2f a, b;
      a.x = sA[lo * 64 + kb + koff];
      a.y = sA[lo * 64 + kb + koff + 1];
      b.x = Wx[(kb + koff) * G4H + n0 + lo];
      b.y = Wx[(kb + koff + 1) * G4H + n0 + lo];
      acc = wmma4(a, b, acc);
    }
    const int rbase = m0 + hi * 8;
#pragma unroll
    for (int v = 0; v < 8; ++v)
      xg[(size_t)(rbase + v) * G4H + n0 + lo] = acc[v];
  }
}

// ---------------------------------------------------------------------------
// Kernel 2: LSTM scan over T=512. 2 workgroups, 16 batch rows each.
// Wh resident in LDS (64 KB of the 320 KB WGP pool); h,c,z in LDS.
// Per step: z = xg[:,t,:] + h @ Wh + bh  (WMMA f32), then gate math (VALU).
// ---------------------------------------------------------------------------
__global__ __launch_bounds__(256) void k_lstm(const float* __restrict__ xg,
                                              const float* __restrict__ Wh,
                                              const float* __restrict__ bh,
                                              float* __restrict__ dec) {
  extern __shared__ float lds[];
  float* sWh = lds;                 // 64*256
  float* sH  = sWh + 64 * G4H;      // 16*64
  float* sC  = sH + 16 * H_;        // 16*64
  float* sZ  = sC + 16 * H_;        // 16*256

  const int b0 = blockIdx.x * 16;
  for (int i = threadIdx.x; i < 64 * G4H; i += 256) sWh[i] = Wh[i];
  for (int i = threadIdx.x; i < 16 * H_; i += 256) { sH[i] = 0.f; sC[i] = 0.f; }
  __syncthreads();

  const int wave = threadIdx.x >> 5;
  const int lane = threadIdx.x & 31;
  const int lo = lane & 15;
  const int hi = lane >> 4;
  const int koff = hi * 2;

  for (int t = 0; t < T_; ++t) {
    // --- z = xg_t + bh + h @ Wh : 8 waves x 2 tiles covers [16,256] ---
    for (int ti = 0; ti < 2; ++ti) {
      const int n0 = (wave * 2 + ti) * 16;
      v8f acc;
#pragma unroll
      for (int v = 0; v < 8; ++v) {
        const int row = v + hi * 8;
        acc[v] = xg[((size_t)(b0 + row) * T_ + t) * G4H + n0 + lo] + bh[n0 + lo];
      }
      for (int kb = 0; kb < H_; kb += 4) {
        v2f a, b;
        a.x = sH[lo * H_ + kb + koff];
        a.y = sH[lo * H_ + kb + koff + 1];
        b.x = sWh[(kb + koff) * G4H + n0 + lo];
        b.y = sWh[(kb + koff + 1) * G4H + n0 + lo];
        acc = wmma4(a, b, acc);
      }
#pragma unroll
      for (int v = 0; v < 8; ++v)
        sZ[(v + hi * 8) * G4H + n0 + lo] = acc[v];
    }
    __syncthreads();

    // --- gate math: 16x64 c/h updates spread over 256 threads ---
#pragma unroll
    for (int e = 0; e < 4; ++e) {
      const int idx = threadIdx.x + e * 256;
      const int row = idx >> 6;
      const int j = idx & 63;
      const float* zr = sZ + row * G4H;
      const float ig = sigf(zr[j]);
      const float fg = sigf(zr[H_ + j]);
      const float gg = tanhf(zr[2 * H_ + j]);
      const float og = sigf(zr[3 * H_ + j]);
      const float c = fg * sC[row * H_ + j] + ig * gg;
      const float h = og * tanhf(c);
      sC[row * H_ + j] = c;
      sH[row * H_ + j] = h;
      dec[((size_t)(b0 + row) * T_ + t) * H_ + j] = h;
    }
    __syncthreads();
  }
}

// ---------------------------------------------------------------------------
// Kernel 3: fused  Y = gelu(layernorm(X @ W + bias) * gamma + beta)
// WG owns 16 rows x all 128 cols so the LN reduction is workgroup-local.
// ---------------------------------------------------------------------------
template <int K>
__global__ __launch_bounds__(256) void k_mlp(const float* __restrict__ X,
                                             const float* __restrict__ W,
                                             const float* __restrict__ bias,
                                             const float* __restrict__ gamma,
                                             const float* __restrict__ beta,
                                             float* __restrict__ Y) {
  __shared__ float sX[16 * K];
  __shared__ float sZ[16 * 128];
  __shared__ float sMu[16];
  __shared__ float sRs[16];

  const int m0 = blockIdx.x * 16;
  for (int i = threadIdx.x; i < 16 * K; i += 256)
    sX[i] = X[(size_t)(m0 + i / K) * K + (i % K)];
  __syncthreads();

  const int wave = threadIdx.x >> 5;
  const int lane = threadIdx.x & 31;
  const int lo = lane & 15;
  const int hi = lane >> 4;
  const int koff = hi * 2;
  const int n0 = wave * 16;

  v8f acc = {};
  for (int kb = 0; kb < K; kb += 4) {
    v2f a, b;
    a.x = sX[lo * K + kb + koff];
    a.y = sX[lo * K + kb + koff + 1];
    b.x = W[(kb + koff) * 128 + n0 + lo];
    b.y = W[(kb + koff + 1) * 128 + n0 + lo];
    acc = wmma4(a, b, acc);
  }
#pragma unroll
  for (int v = 0; v < 8; ++v)
    sZ[(v + hi * 8) * 128 + n0 + lo] = acc[v] + bias[n0 + lo];
  __syncthreads();

  if (threadIdx.x < 16) {
    float s = 0.f, ss = 0.f;
    const float* zr = sZ + threadIdx.x * 128;
    for (int j = 0; j < 128; ++j) { const float z = zr[j]; s += z; ss += z * z; }
    const float mu = s * (1.0f / 128.0f);
    const float var = ss * (1.0f / 128.0f) - mu * mu;
    sMu[threadIdx.x] = mu;
    sRs[threadIdx.x] = rsqrtf(var + 1e-6f);
  }
  __syncthreads();

#pragma unroll
  for (int e = 0; e < 8; ++e) {
    const int idx = threadIdx.x + e * 256;
    const int row = idx >> 7, col = idx & 127;
    const float z = (sZ[idx] - sMu[row]) * sRs[row] * gamma[col] + beta[col];
    Y[(size_t)(m0 + row) * 128 + col] = gelu_tanh(z);
  }
}

// ---------------------------------------------------------------------------
// Kernel 4: out = Y2 @ Wo + bo   [16384,128] x [128,4096]  (256 MB store-bound)
// grid = (32 col-groups of 128, 1024 row tiles); X tile shared in LDS.
// ---------------------------------------------------------------------------
__global__ __launch_bounds__(256) void k_out(const float* __restrict__ X,
                                             const float* __restrict__ Wo,
                                             const float* __restrict__ bo,
                                             float* __restrict__ out) {
  __shared__ float sX[16 * 128];
  const int m0 = blockIdx.y * 16;
  const int g0 = blockIdx.x * 128;
  for (int i = threadIdx.x; i < 16 * 128; i += 256)
    sX[i] = X[(size_t)(m0 + (i >> 7)) * 128 + (i & 127)];
  __syncthreads();

  const int wave = threadIdx.x >> 5;
  const int lane = threadIdx.x & 31;
  const int lo = lane & 15;
  const int hi = lane >> 4;
  const int koff = hi * 2;
  const int n0 = g0 + wave * 16;

  v8f acc = {};
  for (int kb = 0; kb < 128; kb += 4) {
    v2f a, b;
    a.x = sX[lo * 128 + kb + koff];
    a.y = sX[lo * 128 + kb + koff + 1];
    b.x = Wo[(size_t)(kb + koff) * S_ + n0 + lo];
    b.y = Wo[(size_t)(kb + koff + 1) * S_ + n0 + lo];
    acc = wmma4(a, b, acc);
  }
  const float bb = bo[n0 + lo];
  const int rbase = m0 + hi * 8;
#pragma unroll
  for (int v = 0; v < 8; ++v)
    out[(size_t)(rbase + v) * S_ + n0 + lo] = acc[v] + bb;
}

// ---------------------------------------------------------------------------
extern "C" void kernel_launch(void* const* d_in, const int* in_sizes, int n_in,
                              void* d_out, int out_size, void* d_ws, size_t ws_size,
                              hipStream_t stream) {
  const float* enc = (const float*)d_in[0];
  const float* Wx  = (const float*)d_in[1];
  const float* Wh  = (const float*)d_in[2];
  const float* bh  = (const float*)d_in[3];
  const float* W1  = (const float*)d_in[4];
  const float* b1  = (const float*)d_in[5];
  const float* g1  = (const float*)d_in[6];
  const float* be1 = (const float*)d_in[7];
  const float* W2  = (const float*)d_in[8];
  const float* b2  = (const float*)d_in[9];
  const float* g2  = (const float*)d_in[10];
  const float* be2 = (const float*)d_in[11];
  const float* Wo  = (const float*)d_in[12];
  const float* bo  = (const float*)d_in[13];

  float* ws  = (float*)d_ws;
  float* xg  = ws;                          // BT * 256
  float* dec = xg + (size_t)BT * G4H;       // BT * 64
  float* y1  = dec + (size_t)BT * H_;       // BT * 128
  float* y2  = y1 + (size_t)BT * 128;       // BT * 128  (total ~36 MB)
  float* out = (float*)d_out;

  k_xgemm<<<dim3(BT / 16), dim3(256), 0, stream>>>(enc, Wx, xg);

  const size_t lstm_lds =
      (size_t)(64 * G4H + 2 * 16 * H_ + 16 * G4H) * sizeof(float); // ~88 KB
  k_lstm<<<dim3(2), dim3(256), lstm_lds, stream>>>(xg, Wh, bh, dec);

  k_mlp<64><<<dim3(BT / 16), dim3(256), 0, stream>>>(dec, W1, b1, g1, be1, y1);
  k_mlp<128><<<dim3(BT / 16), dim3(256), 0, stream>>>(y1, W2, b2, g2, be2, y2);

  k_out<<<dim3(S_ / 128, BT / 16), dim3(256), 0, stream>>>(y2, Wo, bo, out);
}